// CategoricalAwareTabularEncoder_1460288880936
// MI455X (gfx1250) — compile-verified
//
#include <hip/hip_runtime.h>

typedef float v2f __attribute__((ext_vector_type(2)));
typedef float v8f __attribute__((ext_vector_type(8)));
typedef unsigned int u32x4 __attribute__((ext_vector_type(4)));
typedef int i32x4 __attribute__((ext_vector_type(4)));
typedef int i32x8 __attribute__((ext_vector_type(8)));

#define LD  66           // LDS row stride for activations (floats)
#define LW  65           // LDS row stride for TDM-staged weights (64 + 1 pad DWORD)
#define NPAIRS 2016.0f   // S*(S-1)/2 for S=64

__device__ __forceinline__ v8f wmma4(v2f a, v2f b, v8f c) {
  // D = A(16x4 f32) * B(4x16 f32) + C(16x16 f32)
  return __builtin_amdgcn_wmma_f32_16x16x4_f32(false, a, false, b, (short)0, c,
                                               false, false);
}

// TDM: DMA a [rows x 64] fp32 row-major tile from global into LDS, inserting
// one pad DWORD after every 64 DWORDs (=> LDS row stride 65 floats).
__device__ __forceinline__ void tdm_load_rows64(unsigned lds_addr,
                                                const float* gptr,
                                                unsigned rows) {
  unsigned long long ga = (unsigned long long)(uintptr_t)gptr;
  u32x4 g0;
  g0.x = 0x1u;                                            // count=1, user mode
  g0.y = lds_addr;                                        // LDS byte address
  g0.z = (unsigned)ga;                                    // global_addr[31:0]
  g0.w = ((unsigned)(ga >> 32) & 0x01FFFFFFu) | 0x80000000u; // addr[56:32] | type=2
  i32x8 g1;
  g1[0] = 0x01520000;          // data_size=4B, pad_enable, pad_interval=64DW, pad=1DW
  g1[1] = (int)(64u << 16);    // tensor_dim0 = 64 (bits [79:48] low half)
  g1[2] = (int)(rows << 16);   // tensor_dim0 hi=0 | tensor_dim1 lo16 = rows
  g1[3] = (int)(64u << 16);    // tensor_dim1 hi=0 | tile_dim0 = 64
  g1[4] = (int)rows;           // tile_dim1 = rows, tile_dim2 = 0
  g1[5] = 64;                  // tensor_dim0_stride = 64 (low 32)
  g1[6] = 0;                   // stride0 hi / stride1 lo
  g1[7] = 0;
  i32x4 z4 = {0, 0, 0, 0};
#if defined(__clang_major__) && (__clang_major__ >= 23)
  i32x8 z8 = {0, 0, 0, 0, 0, 0, 0, 0};
  __builtin_amdgcn_tensor_load_to_lds(g0, g1, z4, z4, z8, 0);
#else
  __builtin_amdgcn_tensor_load_to_lds(g0, g1, z4, z4, 0);
#endif
}

__global__ __launch_bounds__(128) void encoder_kernel(
    const int* __restrict__ ids,            // [256,64]
    const float* __restrict__ emb_table,    // [100000,64]
    const float* __restrict__ comb_w1,      // [128,64]
    const float* __restrict__ comb_b1,      // [64]
    const float* __restrict__ comb_w2,      // [64,10]
    const float* __restrict__ comb_b2,      // [10]
    const float* __restrict__ freq_w1,      // [65,64]
    const float* __restrict__ freq_b1,      // [64]
    const float* __restrict__ freq_w2,      // [64,10]
    const float* __restrict__ freq_b2,      // [10]
    const float* __restrict__ centers,      // [10,64]
    const float* __restrict__ cat_freq,     // [100000]
    const float* __restrict__ total_samples,// [1]
    float* __restrict__ out)                // [256,64,10]
{
  __shared__ __attribute__((aligned(16))) float s_w1[128 * LW];  // comb_w1 via TDM
  __shared__ __attribute__((aligned(16))) float s_fw1[65 * LW];  // freq_w1 via TDM
  __shared__ float s_emb[64 * LD];
  __shared__ float s_U[64 * LD];
  __shared__ float s_V[64 * LD];
  __shared__ float s_fh[4][16 * LD];
  __shared__ float s_clus[64 * 16];
  __shared__ float s_ff[64 * 10];
  __shared__ float s_cf[64 * 10];
  __shared__ float s_freqs[64];
  __shared__ float s_c2[16];
  __shared__ float s_Hsum[64];
  __shared__ float s_combo[16];

  const int b    = blockIdx.x;
  const int t    = threadIdx.x;
  const int wave = t >> 5;
  const int lane = t & 31;
  const int half = lane >> 4;     // 0: lanes 0-15, 1: lanes 16-31
  const int l16  = lane & 15;

  // ---- Kick off weight DMA (wave 0) so it overlaps the embedding gather ----
  if (wave == 0) {
    tdm_load_rows64((unsigned)(uintptr_t)s_w1,  comb_w1, 128);
    tdm_load_rows64((unsigned)(uintptr_t)s_fw1, freq_w1, 65);
  }

  // ---------------- Phase 0: gather embeddings + small init ----------------
  {
    int tok = t >> 1;
    int h32 = t & 1;
    int id  = ids[b * 64 + tok];
    const float* src = emb_table + (long)id * 64 + h32 * 32;
    float* dst = &s_emb[tok * LD + h32 * 32];
#pragma unroll
    for (int c = 0; c < 32; c += 4) {
      float4 v = *(const float4*)(src + c);
      dst[c + 0] = v.x; dst[c + 1] = v.y; dst[c + 2] = v.z; dst[c + 3] = v.w;
    }
    if (h32 == 0) s_freqs[tok] = cat_freq[id] / total_samples[0];
    if (t < 64) s_Hsum[t] = 0.0f;
    if (t < 16) {
      float c2 = 0.0f;
      int n = (t < 10) ? t : 9;
      for (int d = 0; d < 64; ++d) { float v = centers[n * 64 + d]; c2 += v * v; }
      s_c2[t] = (t < 10) ? c2 : 0.0f;
    }
  }
  if (wave == 0) __builtin_amdgcn_s_wait_tensorcnt(0);
  __syncthreads();

  // ---------------- Phase 1: per-wave 16-token WMMA GEMMs ----------------
  const int tokbase = wave * 16;

  // A fragments (16x4 f32 tiles over K=64): lane l16 = row, half selects K+2
  v2f a[16];
#pragma unroll
  for (int kt = 0; kt < 16; ++kt) {
    const float* p = &s_emb[(tokbase + l16) * LD + kt * 4 + half * 2];
    a[kt] = *(const v2f*)p;
  }

  // U = emb @ comb_w1[0:64,:]  and  V = emb @ comb_w1[64:128,:]  (B from LDS)
#pragma unroll
  for (int nt = 0; nt < 4; ++nt) {
    const int n = nt * 16 + l16;
    v8f cu = {}; v8f cv = {};
#pragma unroll
    for (int kt = 0; kt < 16; ++kt) {
      const int k0 = kt * 4 + half * 2;
      v2f bu, bv;
      bu.x = s_w1[(k0 + 0) * LW + n];
      bu.y = s_w1[(k0 + 1) * LW + n];
      bv.x = s_w1[(k0 + 64) * LW + n];
      bv.y = s_w1[(k0 + 65) * LW + n];
      cu = wmma4(a[kt], bu, cu);
      cv = wmma4(a[kt], bv, cv);
    }
#pragma unroll
    for (int r = 0; r < 8; ++r) {
      const int m = r + half * 8;
      s_U[(tokbase + m) * LD + n] = cu[r];
      s_V[(tokbase + m) * LD + n] = cv[r];
    }
  }

  // freq hidden: relu(emb @ freq_w1[0:64,:] + freqs*freq_w1[64,:] + freq_b1)
#pragma unroll
  for (int nt = 0; nt < 4; ++nt) {
    const int n = nt * 16 + l16;
    v8f c = {};
#pragma unroll
    for (int kt = 0; kt < 16; ++kt) {
      const int k0 = kt * 4 + half * 2;
      v2f bb;
      bb.x = s_fw1[(k0 + 0) * LW + n];
      bb.y = s_fw1[(k0 + 1) * LW + n];
      c = wmma4(a[kt], bb, c);
    }
    const float w1last = s_fw1[64 * LW + n];
    const float fb1    = freq_b1[n];
#pragma unroll
    for (int r = 0; r < 8; ++r) {
      const int m = r + half * 8;
      float v = c[r] + s_freqs[tokbase + m] * w1last + fb1;
      s_fh[wave][m * LD + n] = fmaxf(v, 0.0f);
    }
  }

  // cluster logits: emb @ centers^T  (N padded 10 -> 16 with zeros)
  {
    const int n  = l16;
    const int nc = (n < 10) ? n : 9;
    const float msk = (n < 10) ? 1.0f : 0.0f;
    v8f c = {};
#pragma unroll
    for (int kt = 0; kt < 16; ++kt) {
      const int k0 = kt * 4 + half * 2;
      v2f bb;
      bb.x = centers[nc * 64 + k0 + 0] * msk;
      bb.y = centers[nc * 64 + k0 + 1] * msk;
      c = wmma4(a[kt], bb, c);
    }
#pragma unroll
    for (int r = 0; r < 8; ++r)
      s_clus[(tokbase + r + half * 8) * 16 + n] = c[r];
  }

  // freq output: fh @ freq_w2 + freq_b2  (A re-staged from LDS, N pad 10->16)
  {
    v2f af[16];
#pragma unroll
    for (int kt = 0; kt < 16; ++kt)
      af[kt] = *(const v2f*)&s_fh[wave][l16 * LD + kt * 4 + half * 2];

    const int n  = l16;
    const int nc = (n < 10) ? n : 9;
    const float msk = (n < 10) ? 1.0f : 0.0f;
    v8f c = {};
#pragma unroll
    for (int kt = 0; kt < 16; ++kt) {
      const int k0 = kt * 4 + half * 2;
      v2f bb;
      bb.x = freq_w2[(k0 + 0) * 10 + nc] * msk;
      bb.y = freq_w2[(k0 + 1) * 10 + nc] * msk;
      c = wmma4(af[kt], bb, c);
    }
    if (n < 10) {
      const float fb2 = freq_b2[n];
#pragma unroll
      for (int r = 0; r < 8; ++r)
        s_ff[(tokbase + r + half * 8) * 10 + n] = c[r] + fb2;
    }
  }
  __syncthreads();

  // ---------------- Phase 2: pair accumulation Hsum[d] ----------------
  {
    const int d = t & 63;
    const int h = t >> 6;           // split j's odd/even across thread halves
    const float b1 = comb_b1[d];
    float acc = 0.0f;
    for (int j = 1 + h; j < 64; j += 2) {
      const float vj = s_V[j * LD + d] + b1;
      for (int i = 0; i < j; ++i)
        acc += fmaxf(s_U[i * LD + d] + vj, 0.0f);
    }
    atomicAdd(&s_Hsum[d], acc);     // ds_add_f32
  }
  __syncthreads();

  // ---------------- Phase 3: combo_sum + cluster softmax ----------------
  if (t < 10) {
    float acc = NPAIRS * comb_b2[t];
    for (int d = 0; d < 64; ++d) acc += s_Hsum[d] * comb_w2[d * 10 + t];
    s_combo[t] = acc;
  }
  if (t < 64) {
    float e2 = 0.0f;
    for (int d = 0; d < 64; ++d) { float v = s_emb[t * LD + d]; e2 += v * v; }
    float lg[10];
    float mx = -1e30f;
#pragma unroll
    for (int n = 0; n < 10; ++n) {
      float d2   = e2 + s_c2[n] - 2.0f * s_clus[t * 16 + n];
      float dist = sqrtf(fmaxf(d2, 0.0f));
      lg[n] = -dist;
      mx = fmaxf(mx, lg[n]);
    }
    float sum = 0.0f;
#pragma unroll
    for (int n = 0; n < 10; ++n) { lg[n] = __expf(lg[n] - mx); sum += lg[n]; }
    const float inv = 1.0f / sum;
#pragma unroll
    for (int n = 0; n < 10; ++n) s_cf[t * 10 + n] = lg[n] * inv;
  }
  __syncthreads();

  // ---------------- Phase 4: final mix + store ----------------
  {
    const float inv = 1.0f / (NPAIRS + 2.0f);
    for (int idx = t; idx < 640; idx += 128) {
      const int f = idx % 10;
      out[b * 640 + idx] = (s_ff[idx] + s_combo[f] + s_cf[idx]) * inv;
    }
  }
}

extern "C" void kernel_launch(void* const* d_in, const int* in_sizes, int n_in,
                              void* d_out, int out_size, void* d_ws, size_t ws_size,
                              hipStream_t stream) {
  (void)in_sizes; (void)n_in; (void)out_size; (void)d_ws; (void)ws_size;
  encoder_kernel<<<256, 128, 0, stream>>>(
      (const int*)d_in[0],    // category_ids
      (const float*)d_in[1],  // emb_table
      (const float*)d_in[2],  // comb_w1
      (const float*)d_in[3],  // comb_b1
      (const float*)d_in[4],  // comb_w2
      (const float*)d_in[5],  // comb_b2
      (const float*)d_in[6],  // freq_w1
      (const float*)d_in[7],  // freq_b1
      (const float*)d_in[8],  // freq_w2
      (const float*)d_in[9],  // freq_b2
      (const float*)d_in[10], // cluster_centers
      (const float*)d_in[11], // cat_freq
      (const float*)d_in[12], // total_samples
      (float*)d_out);
}